// MultiHeadAttention_35167192220038
// MI455X (gfx1250) — compile-verified
//
#include <hip/hip_runtime.h>
#include <math.h>

// MI455X / gfx1250: wave32, WMMA (fp32 path: v_wmma_f32_16x16x4_f32).
// Pipeline: qkv projection -> flash causal attention -> residual + custom LayerNorm.

typedef __attribute__((ext_vector_type(2))) float v2f;
typedef __attribute__((ext_vector_type(8))) float v8f;

constexpr int Bc = 2, Sc = 2048, Ec = 1024, Hc = 16, Dc = 64;
constexpr float EPSc = 1e-6f;

#define V8F_ZERO {0.f,0.f,0.f,0.f,0.f,0.f,0.f,0.f}

static __device__ __forceinline__ v8f wmma_f32_k4(v2f a, v2f b, v8f c) {
  // D = A(16x4) * B(4x16) + C(16x16), all fp32
  return __builtin_amdgcn_wmma_f32_16x16x4_f32(
      /*neg_a=*/false, a, /*neg_b=*/false, b,
      /*c_mod=*/(short)0, c, /*reuse_a=*/false, /*reuse_b=*/false);
}

// ---------------------------------------------------------------------------
// Kernel 1: per-head projection  Y[b,h,s,d] = X[b,s,:] @ W[h,:,d] + bias[h,d]
// One wave per 16-row M tile; computes all four 16-col D tiles (A reuse 4x).
// ---------------------------------------------------------------------------
__global__ void __launch_bounds__(128) qkv_proj_kernel(
    const float* __restrict__ X, const float* __restrict__ W,
    const float* __restrict__ bias, float* __restrict__ Y) {
  const int wave = threadIdx.x >> 5;
  const int lane = threadIdx.x & 31;
  const int g    = lane >> 4;     // half-wave group (K split for A/B operands)
  const int ln   = lane & 15;
  const int bh   = blockIdx.y;
  const int b    = bh / Hc, h = bh % Hc;
  const int m0   = (blockIdx.x * 4 + wave) * 16;

  const float* Xrow = X + ((size_t)b * Sc + m0 + ln) * Ec;  // A: m = ln
  const float* Wh   = W + (size_t)h * Ec * Dc;

  v8f c0 = V8F_ZERO, c1 = V8F_ZERO, c2 = V8F_ZERO, c3 = V8F_ZERO;
  {
    const float* bp = bias + h * Dc;
    const float b0 = bp[ 0 + ln], b1 = bp[16 + ln], b2 = bp[32 + ln], b3 = bp[48 + ln];
#pragma unroll
    for (int i = 0; i < 8; ++i) { c0[i] = b0; c1[i] = b1; c2[i] = b2; c3[i] = b3; }
  }

#pragma unroll 4
  for (int ko = 0; ko < Ec; ko += 4) {
    const int k = ko + 2 * g;
    v2f a; a.x = Xrow[k]; a.y = Xrow[k + 1];
    const float* wr0 = Wh + (size_t)k * Dc + ln;
    const float* wr1 = wr0 + Dc;
    v2f bb;
    bb.x = wr0[ 0]; bb.y = wr1[ 0]; c0 = wmma_f32_k4(a, bb, c0);
    bb.x = wr0[16]; bb.y = wr1[16]; c1 = wmma_f32_k4(a, bb, c1);
    bb.x = wr0[32]; bb.y = wr1[32]; c2 = wmma_f32_k4(a, bb, c2);
    bb.x = wr0[48]; bb.y = wr1[48]; c3 = wmma_f32_k4(a, bb, c3);
  }

  // C/D tile: lane holds col=ln, rows g*8+i
  float* Yb = Y + (((size_t)b * Hc + h) * Sc + m0 + g * 8) * Dc + ln;
#pragma unroll
  for (int i = 0; i < 8; ++i) {
    Yb[i * Dc +  0] = c0[i];
    Yb[i * Dc + 16] = c1[i];
    Yb[i * Dc + 32] = c2[i];
    Yb[i * Dc + 48] = c3[i];
  }
}

// ---------------------------------------------------------------------------
// Kernel 2: causal flash attention per (b,h). One wave owns a 16x64 out tile.
// ---------------------------------------------------------------------------
__global__ void __launch_bounds__(128) attn_kernel(
    const float* __restrict__ Q, const float* __restrict__ K,
    const float* __restrict__ V, float* __restrict__ ctx) {
  const int wave = threadIdx.x >> 5;
  const int lane = threadIdx.x & 31;
  const int g    = lane >> 4;
  const int ln   = lane & 15;
  const int bh   = blockIdx.y;                 // b*H + h
  const int b    = bh / Hc, h = bh % Hc;
  const int mt   = blockIdx.x * 4 + wave;      // query tile index
  const int m0   = mt * 16;

  __shared__ float pbuf[4][16][17];            // stride-17 pad: conflict-free transpose

  const float* Qh = Q + (size_t)bh * Sc * Dc;
  const float* Kh = K + (size_t)bh * Sc * Dc;
  const float* Vh = V + (size_t)bh * Sc * Dc;

  v8f o0 = V8F_ZERO, o1 = V8F_ZERO, o2 = V8F_ZERO, o3 = V8F_ZERO;
  float rmax[8], rsum[8];
#pragma unroll
  for (int i = 0; i < 8; ++i) { rmax[i] = -INFINITY; rsum[i] = 0.f; }

  const float* Qr = Qh + (size_t)(m0 + ln) * Dc;

  for (int t = 0; t <= mt; ++t) {
    if (t + 1 <= mt) __builtin_prefetch(Kh + (size_t)((t + 1) * 16 + ln) * Dc, 0, 0);

    // ---- scores S = Q Kt^T  (16x16 tile, K-depth 64) ----
    const float* Kr = Kh + (size_t)(t * 16 + ln) * Dc;
    v8f sc = V8F_ZERO;
#pragma unroll
    for (int ko = 0; ko < Dc; ko += 4) {
      const int k = ko + 2 * g;
      v2f a;  a.x  = Qr[k]; a.y  = Qr[k + 1];
      v2f bb; bb.x = Kr[k]; bb.y = Kr[k + 1];
      sc = wmma_f32_k4(a, bb, sc);
    }

    // scale 1/sqrt(64); causal mask only on the diagonal tile
    const bool diag = (t == mt);
#pragma unroll
    for (int i = 0; i < 8; ++i) {
      float s = sc[i] * 0.125f;
      if (diag && ln > g * 8 + i) s = -INFINITY;
      sc[i] = s;
    }

    // ---- online softmax; row r=g*8+i lives across the 16 lanes of group g ----
#pragma unroll
    for (int i = 0; i < 8; ++i) {
      float m = sc[i];
#pragma unroll
      for (int off = 1; off < 16; off <<= 1) m = fmaxf(m, __shfl_xor(m, off, 16));
      const float nm = fmaxf(rmax[i], m);
      const float sf = __expf(rmax[i] - nm);   // exp(-inf)=0 on first tile
      rmax[i] = nm;
      const float p = __expf(sc[i] - nm);
      sc[i] = p;
      float ps = p;
#pragma unroll
      for (int off = 1; off < 16; off <<= 1) ps += __shfl_xor(ps, off, 16);
      rsum[i] = rsum[i] * sf + ps;
      o0[i] *= sf; o1[i] *= sf; o2[i] *= sf; o3[i] *= sf;
    }

    // ---- transpose P: C-layout -> row-major LDS -> A-layout ----
#pragma unroll
    for (int i = 0; i < 8; ++i) pbuf[wave][g * 8 + i][ln] = sc[i];
    asm volatile("s_wait_dscnt 0x0" ::: "memory");

    // ---- O += P(16x16) @ V(16x64) ----
    const float* Vt = Vh + (size_t)(t * 16) * Dc;
#pragma unroll
    for (int ko = 0; ko < 16; ko += 4) {
      const int k = ko + 2 * g;
      v2f a; a.x = pbuf[wave][ln][k]; a.y = pbuf[wave][ln][k + 1];
      const float* vr0 = Vt + (size_t)k * Dc + ln;
      const float* vr1 = vr0 + Dc;
      v2f bb;
      bb.x = vr0[ 0]; bb.y = vr1[ 0]; o0 = wmma_f32_k4(a, bb, o0);
      bb.x = vr0[16]; bb.y = vr1[16]; o1 = wmma_f32_k4(a, bb, o1);
      bb.x = vr0[32]; bb.y = vr1[32]; o2 = wmma_f32_k4(a, bb, o2);
      bb.x = vr0[48]; bb.y = vr1[48]; o3 = wmma_f32_k4(a, bb, o3);
    }
  }

  // normalize rows and store concat-head ctx[b, s, h*64 + d]
  float* crow = ctx + ((size_t)b * Sc + m0 + g * 8) * Ec + h * Dc + ln;
#pragma unroll
  for (int i = 0; i < 8; ++i) {
    const float inv = 1.0f / rsum[i];
    crow[i * Ec +  0] = o0[i] * inv;
    crow[i * Ec + 16] = o1[i] * inv;
    crow[i * Ec + 32] = o2[i] * inv;
    crow[i * Ec + 48] = o3[i] * inv;
  }
}

// ---------------------------------------------------------------------------
// Kernel 3: residual add + custom LayerNorm (unbiased std, shift in denom)
// ---------------------------------------------------------------------------
__global__ void __launch_bounds__(256) ln_kernel(
    const float* __restrict__ ctx, const float* __restrict__ resid,
    const float* __restrict__ lnsc, const float* __restrict__ lnsh,
    float* __restrict__ out) {
  const int row = blockIdx.x;
  const int tid = threadIdx.x;
  const float* xr = ctx   + (size_t)row * Ec;
  const float* rr = resid + (size_t)row * Ec;

  float x[4];
  float loc = 0.f;
#pragma unroll
  for (int j = 0; j < 4; ++j) {
    const int e = tid + j * 256;
    x[j] = xr[e] + rr[e];
    loc += x[j];
  }

  __shared__ float red[8];
#pragma unroll
  for (int off = 16; off >= 1; off >>= 1) loc += __shfl_xor(loc, off, 32);
  if ((tid & 31) == 0) red[tid >> 5] = loc;
  __syncthreads();
  float tot = 0.f;
#pragma unroll
  for (int w = 0; w < 8; ++w) tot += red[w];
  const float mean = tot * (1.0f / Ec);

  float ls = 0.f;
#pragma unroll
  for (int j = 0; j < 4; ++j) { const float d = x[j] - mean; ls += d * d; }
  __syncthreads();
#pragma unroll
  for (int off = 16; off >= 1; off >>= 1) ls += __shfl_xor(ls, off, 32);
  if ((tid & 31) == 0) red[tid >> 5] = ls;
  __syncthreads();
  float tot2 = 0.f;
#pragma unroll
  for (int w = 0; w < 8; ++w) tot2 += red[w];
  const float stdv = sqrtf(tot2 * (1.0f / (Ec - 1)));

  float* orow = out + (size_t)row * Ec;
#pragma unroll
  for (int j = 0; j < 4; ++j) {
    const int e = tid + j * 256;
    orow[e] = lnsc[e] * (x[j] - mean) / (stdv + EPSc + lnsh[e]);
  }
}

// ---------------------------------------------------------------------------
extern "C" void kernel_launch(void* const* d_in, const int* in_sizes, int n_in,
                              void* d_out, int out_size, void* d_ws, size_t ws_size,
                              hipStream_t stream) {
  const float* query    = (const float*)d_in[0];
  const float* key_in   = (const float*)d_in[1];
  const float* value_in = (const float*)d_in[2];
  const float* resid    = (const float*)d_in[3];
  const float* Wq = (const float*)d_in[4];
  const float* bq = (const float*)d_in[5];
  const float* Wk = (const float*)d_in[6];
  const float* bk = (const float*)d_in[7];
  const float* Wv = (const float*)d_in[8];
  const float* bv = (const float*)d_in[9];
  const float* lnsc = (const float*)d_in[10];
  const float* lnsh = (const float*)d_in[11];
  float* out = (float*)d_out;

  const size_t per = (size_t)Bc * Hc * Sc * Dc;  // 4,194,304 floats each
  float* qws = (float*)d_ws;
  float* kws = qws + per;
  float* vws = kws + per;
  float* ctx = vws + per;                        // [B,S,E], total ws = 64 MiB

  dim3 gp(Sc / 64, Bc * Hc);
  qkv_proj_kernel<<<gp, 128, 0, stream>>>(query,    Wq, bq, qws);
  qkv_proj_kernel<<<gp, 128, 0, stream>>>(key_in,   Wk, bk, kws);
  qkv_proj_kernel<<<gp, 128, 0, stream>>>(value_in, Wv, bv, vws);
  attn_kernel<<<dim3(Sc / 64, Bc * Hc), 128, 0, stream>>>(qws, kws, vws, ctx);
  ln_kernel<<<Bc * Sc, 256, 0, stream>>>(ctx, resid, lnsc, lnsh, out);

  (void)in_sizes; (void)n_in; (void)out_size; (void)ws_size;
}